// GCN_81587198755341
// MI455X (gfx1250) — compile-verified
//
#include <hip/hip_runtime.h>
#include <hip/hip_bf16.h>

typedef __attribute__((ext_vector_type(2))) float v2f;
typedef __attribute__((ext_vector_type(4))) float v4f;
typedef __attribute__((ext_vector_type(8))) float v8f;

#define IN_CH 256
#define HID   16

// ---------------- degree / normalization ----------------

__global__ void k_deg_init(unsigned* __restrict__ deg, int N) {
    int n = blockIdx.x * blockDim.x + threadIdx.x;
    if (n < N) deg[n] = 1u;                 // self-loop contributes 1
}

__global__ void k_deg_count(const int* __restrict__ ei, int E,
                            unsigned* __restrict__ deg) {
    int e = blockIdx.x * blockDim.x + threadIdx.x;
    if (e < E) {
        int t = __builtin_nontemporal_load(ei + E + e);   // targets = ei[1][e]
        atomicAdd(&deg[t], 1u);
    }
}

__global__ void k_dinv(const unsigned* __restrict__ deg,
                       float* __restrict__ dinv, int N) {
    int n = blockIdx.x * blockDim.x + threadIdx.x;
    if (n < N) dinv[n] = rsqrtf((float)deg[n]);  // deg >= 1 always
}

// ---------------- h = x @ conv_w  (WMMA f32 16x16x4) ----------------
// Block = 64 threads = 2 waves; each wave computes a 16(node) x 16(hid) tile.
#define XPAD 4   // LDS row padding: stride 260 % 64 == 4 -> conflict-free

__global__ __launch_bounds__(64)
void k_gemm_wmma(const float* __restrict__ x,
                 const float* __restrict__ w,     // [256][16] row-major
                 float* __restrict__ h, int N) {
    __shared__ float sW[IN_CH][HID];              // 16 KB
    __shared__ float sX[2][16][IN_CH + XPAD];     // ~33.3 KB

    const int tid  = threadIdx.x;
    const int wv   = tid >> 5;                    // wave id 0..1
    const int lane = tid & 31;
    const int l16  = lane & 15;
    const int half = lane >> 4;                   // 0: K pair {k,k+1}, 1: {k+2,k+3}
    const int node_base = blockIdx.x * 32 + wv * 16;
    const bool full_tile = (node_base + 16 <= N);

    // cooperative load of conv_w (flat v4f copy, coalesced)
    {
        const v4f* ws = (const v4f*)w;
        v4f* wd = (v4f*)&sW[0][0];
#pragma unroll
        for (int i = 0; i < (IN_CH * HID) / 4 / 64; ++i)   // 16 iters
            wd[tid + i * 64] = ws[tid + i * 64];
    }

    // stage this wave's 16x256 x-tile
    if (full_tile) {
        const v4f* xs = (const v4f*)(x + (size_t)node_base * IN_CH);
#pragma unroll
        for (int it = 0; it < 32; ++it) {
            int fi = it * 32 + lane;              // float4 index within tile
            v4f v = __builtin_nontemporal_load(xs + fi);
            int f = fi * 4;
            *(v4f*)&sX[wv][f >> 8][f & 255] = v;
        }
    } else {
        for (int it = 0; it < 32; ++it) {
            int f = it * 128 + lane * 4;
            int r = f >> 8, c = f & 255;
            int row = node_base + r;
            v4f v = {};
            if (row < N) v = *(const v4f*)(x + (size_t)row * IN_CH + c);
            *(v4f*)&sX[wv][r][c] = v;
        }
    }
    __syncthreads();

    v8f acc = {};
#pragma unroll 8
    for (int k = 0; k < IN_CH; k += 4) {
        // A 16x4: lanes 0-15 -> row l16, K = {k,k+1}; lanes 16-31 -> K = {k+2,k+3}
        v2f a;
        a.x = sX[wv][l16][k + 2 * half + 0];
        a.y = sX[wv][l16][k + 2 * half + 1];
        // B 4x16: VGPR0 = rows {k, k+2}, VGPR1 = rows {k+1, k+3}, N = l16
        v2f b;
        b.x = sW[k + 2 * half + 0][l16];
        b.y = sW[k + 2 * half + 1][l16];
        acc = __builtin_amdgcn_wmma_f32_16x16x4_f32(
            false, a, false, b, (short)0, acc, false, false);
    }

    // C/D layout: VGPR i holds rows (i, i+8), column = l16
    if (full_tile) {
        unsigned base = (unsigned)(node_base + 8 * half) * HID + (unsigned)l16;
#pragma unroll
        for (int i = 0; i < 8; ++i)
            h[base + (unsigned)i * HID] = acc[i];
    } else {
#pragma unroll
        for (int i = 0; i < 8; ++i) {
            int row = node_base + i + 8 * half;
            if (row < N) h[(size_t)row * HID + l16] = acc[i];
        }
    }
}

// ---------------- self-loop init: agg = h * dinv^2 ----------------

__global__ void k_self_init(const float* __restrict__ h,
                            const float* __restrict__ dinv,
                            float* __restrict__ agg, int N) {
    int n = blockIdx.x * blockDim.x + threadIdx.x;
    if (n >= N) return;
    float c = dinv[n];
    c *= c;
    const v4f* hv = (const v4f*)(h + (size_t)n * HID);
    v4f* av = (v4f*)(agg + (size_t)n * HID);
#pragma unroll
    for (int i = 0; i < 4; ++i)
        av[i] = hv[i] * c;
}

// ---------------- edge scatter: agg[t] += h[s] * dinv[s]*dinv[t] ----------------
// one thread per (edge, channel); h/agg/dinv stay L2-resident (~13 MB total)

__global__ void k_edge_scatter(const int* __restrict__ ei, int E,
                               const float* __restrict__ dinv,
                               const float* __restrict__ h,
                               float* __restrict__ agg) {
    long long gid = (long long)blockIdx.x * blockDim.x + threadIdx.x;
    int e = (int)(gid >> 4);
    int j = (int)(gid & 15);
    if (e >= E) return;
    int s = __builtin_nontemporal_load(ei + e);
    int t = __builtin_nontemporal_load(ei + E + e);
    float coef = dinv[s] * dinv[t];
    unsafeAtomicAdd(&agg[(size_t)t * HID + j], h[(size_t)s * HID + j] * coef);
}

// ---------------- bias + relu + out projection ----------------

__global__ void k_out(const float* __restrict__ agg,
                      const float* __restrict__ conv_b,
                      const float* __restrict__ out_w,
                      const float* __restrict__ out_b,
                      float* __restrict__ out, int N) {
    int n = blockIdx.x * blockDim.x + threadIdx.x;
    if (n >= N) return;
    const v4f* av = (const v4f*)(agg + (size_t)n * HID);
    const v4f* bv = (const v4f*)conv_b;
    const v4f* wv = (const v4f*)out_w;
    float acc = out_b[0];
#pragma unroll
    for (int i = 0; i < 4; ++i) {
        v4f v = av[i] + bv[i];
        v4f wvi = wv[i];
#pragma unroll
        for (int j = 0; j < 4; ++j)
            acc = fmaf(fmaxf(v[j], 0.0f), wvi[j], acc);
    }
    out[n] = acc;
}

// ---------------- launch ----------------

static inline size_t align256(size_t v) { return (v + 255) & ~(size_t)255; }

extern "C" void kernel_launch(void* const* d_in, const int* in_sizes, int n_in,
                              void* d_out, int out_size, void* d_ws, size_t ws_size,
                              hipStream_t stream) {
    const float* x      = (const float*)d_in[0];
    const int*   ei     = (const int*)d_in[1];
    const float* conv_w = (const float*)d_in[2];
    const float* conv_b = (const float*)d_in[3];
    const float* out_w  = (const float*)d_in[4];
    const float* out_b  = (const float*)d_in[5];
    float* out = (float*)d_out;

    const int N = in_sizes[0] / IN_CH;
    const int E = in_sizes[1] / 2;

    char* ws = (char*)d_ws;
    size_t off = 0;
    unsigned* deg = (unsigned*)(ws + off); off = align256(off + (size_t)N * 4);
    float* dinv   = (float*)(ws + off);    off = align256(off + (size_t)N * 4);
    float* h      = (float*)(ws + off);    off = align256(off + (size_t)N * HID * 4);
    float* agg    = (float*)(ws + off);    off = align256(off + (size_t)N * HID * 4);

    const int nb_n = (N + 255) / 256;
    const int nb_e = (E + 255) / 256;

    k_deg_init<<<nb_n, 256, 0, stream>>>(deg, N);
    k_deg_count<<<nb_e, 256, 0, stream>>>(ei, E, deg);
    k_dinv<<<nb_n, 256, 0, stream>>>(deg, dinv, N);

    k_gemm_wmma<<<(N + 31) / 32, 64, 0, stream>>>(x, conv_w, h, N);

    k_self_init<<<nb_n, 256, 0, stream>>>(h, dinv, agg, N);

    long long scatter_threads = (long long)E * HID;
    int nb_s = (int)((scatter_threads + 255) / 256);
    k_edge_scatter<<<nb_s, 256, 0, stream>>>(ei, E, dinv, h, agg);

    k_out<<<nb_n, 256, 0, stream>>>(agg, conv_b, out_w, out_b, out, N);
}